// ManifoldAttentionNoAttnStage1_75522704933232
// MI455X (gfx1250) — compile-verified
//
#include <hip/hip_runtime.h>
#include <cmath>
#include <complex>

// ---------------- problem constants ----------------
#define B_ 4
#define T_ 1024
#define D_ 512
#define R_ 8
#define L_ 2048
#define LOGL_ 11
#define PI_F 3.14159265358979323846f
#define EPSM 1e-5f
#define LAM_ 0.02f

typedef __attribute__((ext_vector_type(2))) float v2f;
typedef __attribute__((ext_vector_type(8))) float v8f;
typedef __attribute__((ext_vector_type(4))) int v4i;
typedef __attribute__((ext_vector_type(8))) int v8i;

struct FrftP { float s, cps, prefRe, prefIm; };
struct FrftPack { FrftP p[16]; };   // 0..7 = +alpha, 8..15 = -alpha

// ---------------- device helpers ----------------
__device__ __forceinline__ float2 cmulf(float2 a, float2 b) {
    return make_float2(a.x * b.x - a.y * b.y, a.x * b.y + a.y * b.x);
}

__device__ __forceinline__ float2 prepost_ph(float cps, int t) {
    float tn = -1.0f + 2.0f * (float)t / 1023.0f;
    float ang = PI_F * cps * tn * tn;
    float sn, cs;
    __sincosf(ang, &sn, &cs);
    return make_float2(cs, sn);
}

// in-LDS radix-2 DIT FFT, length 2048. dir = -1 forward, +1 inverse (unscaled).
__device__ void fft2048(float2* s, float dir) {
    int tid = threadIdx.x, nt = blockDim.x;
    __syncthreads();
    for (int i = tid; i < L_; i += nt) {
        int j = (int)(__brev((unsigned)i) >> (32 - LOGL_));
        if (j > i) { float2 t = s[i]; s[i] = s[j]; s[j] = t; }
    }
    __syncthreads();
    for (int len = 2; len <= L_; len <<= 1) {
        int half = len >> 1;
        float ang0 = dir * 2.0f * PI_F / (float)len;
        for (int k = tid; k < (L_ >> 1); k += nt) {
            int grp = k / half, pos = k - grp * half;
            int i0 = grp * len + pos, i1 = i0 + half;
            float sn, cs;
            __sincosf(ang0 * (float)pos, &sn, &cs);
            float2 u = s[i0], v = s[i1];
            float2 wv = make_float2(cs * v.x - sn * v.y, cs * v.y + sn * v.x);
            s[i0] = make_float2(u.x + wv.x, u.y + wv.y);
            s[i1] = make_float2(u.x - wv.x, u.y - wv.y);
        }
        __syncthreads();
    }
}

// f32 WMMA: D(16x16) = A(16x4) * B(4x16) + C
__device__ __forceinline__ v8f wmma4(v2f a, v2f b, v8f c) {
    return __builtin_amdgcn_wmma_f32_16x16x4_f32(false, a, false, b, (short)0, c,
                                                 false, false);
}

// ---------------- kernels ----------------
__global__ void zero_f32(float* p, int n) {
    int i = blockIdx.x * blockDim.x + threadIdx.x;
    if (i < n) p[i] = 0.f;
}

// Build Bluestein chirp spectrum H for each of 16 alpha-sets.
__global__ void build_H(float2* Hout, FrftPack pk) {
    __shared__ float2 buf[L_];
    int ai = blockIdx.x;
    FrftP P = pk.p[ai];
    const float dt = 2.0f / 1023.0f;
    for (int idx = threadIdx.x; idx < L_; idx += blockDim.x) {
        float2 v = make_float2(0.f, 0.f);
        if (idx != 1024) {
            int m = (idx <= 1023) ? idx : idx - L_;
            float md = (float)m * dt;
            float ang = -PI_F * P.s * md * md;
            float sn, cs;
            __sincosf(ang, &sn, &cs);
            v = make_float2(cs, sn);
        }
        buf[idx] = v;
    }
    fft2048(buf, -1.0f);
    for (int idx = threadIdx.x; idx < L_; idx += blockDim.x)
        Hout[(size_t)ai * L_ + idx] = buf[idx];
}

// FrFT of xprime channel (b,d) for alpha ai; accumulate |X|^2 into E[ai][b][t].
__global__ void frft_energy(const float* __restrict__ x, const float2* __restrict__ H,
                            float* __restrict__ E, FrftPack pk) {
    __shared__ float2 buf[L_];
    int blk = blockIdx.x;
    int ai = blk / (B_ * D_);
    int rem = blk % (B_ * D_);
    int b = rem / D_, d = rem % D_;
    FrftP P = pk.p[ai];
    const float dt = 2.0f / 1023.0f;
    float sub = (d == 0) ? 1.0f : 0.0f;
    for (int t = threadIdx.x; t < L_; t += blockDim.x) {
        float2 v = make_float2(0.f, 0.f);
        if (t < T_) {
            float xr = x[((size_t)b * T_ + t) * D_ + d] - sub;
            float2 pp = prepost_ph(P.cps, t);
            v = make_float2(xr * pp.x, xr * pp.y);
        }
        buf[t] = v;
    }
    fft2048(buf, -1.0f);
    const float2* Hc = H + (size_t)ai * L_;
    for (int k = threadIdx.x; k < L_; k += blockDim.x) buf[k] = cmulf(buf[k], Hc[k]);
    fft2048(buf, +1.0f);
    float2 pref = make_float2(P.prefRe, P.prefIm);
    for (int t = threadIdx.x; t < T_; t += blockDim.x) {
        float2 u = buf[1023 + t];
        u.x *= (1.0f / (float)L_); u.y *= (1.0f / (float)L_);
        float2 pp = prepost_ph(P.cps, t);
        float2 val = cmulf(cmulf(pref, pp), u);
        val.x *= dt; val.y *= dt;
        float e = val.x * val.x + val.y * val.y;
        atomicAdd(&E[((size_t)ai * B_ + b) * T_ + t], e);
    }
}

// w = sqrt(E/D + 1e-6); w /= (mean_t(w) + 1e-6)   (in place, one block per (ai,b))
__global__ void finalize_w(float* __restrict__ Ew) {
    __shared__ float red[256];
    float* row = Ew + (size_t)blockIdx.x * T_;
    float part = 0.f;
    for (int t = threadIdx.x; t < T_; t += blockDim.x) {
        float w = sqrtf(row[t] * (1.0f / (float)D_) + 1e-6f);
        row[t] = w; part += w;
    }
    red[threadIdx.x] = part;
    __syncthreads();
    for (int s = 128; s > 0; s >>= 1) {
        if (threadIdx.x < s) red[threadIdx.x] += red[threadIdx.x + s];
        __syncthreads();
    }
    float inv = 1.0f / (red[0] / (float)T_ + 1e-6f);
    for (int t = threadIdx.x; t < T_; t += blockDim.x) row[t] *= inv;
}

__global__ void init_V(float* V) {
    int i = blockIdx.x * blockDim.x + threadIdx.x;
    if (i < B_ * D_ * R_) {
        int j = i % R_, d = (i / R_) % D_;
        V[i] = (d == j) ? 1.0f : 0.0f;
    }
}

// Y[b,t,j] = sum_d xprime[b,t,d] * V[b,d,j]   (one wave per 16-row M-tile; WMMA f32)
__global__ void gemm_xprime_V(const float* __restrict__ x, const float* __restrict__ V,
                              float* __restrict__ Y) {
    int b = blockIdx.x / (T_ / 16), mt = blockIdx.x % (T_ / 16);
    int lane = threadIdx.x;
    int mrow = lane & 15;
    int koff = (lane >> 4) << 1;   // 0 or 2
    int n = lane & 15;
    int nn = n & 7;                         // safe padded column
    float bmask = (n < R_) ? 1.0f : 0.0f;   // zero out padded N columns
    const float2* A2 = (const float2*)(x + ((size_t)b * T_ + (size_t)mt * 16) * D_);
    const float* Vb = V + (size_t)b * D_ * R_;
    v8f c = {0.f, 0.f, 0.f, 0.f, 0.f, 0.f, 0.f, 0.f};
    {   // peeled kk == 0: apply the xprime anchor (column d==0) without a loop branch
        int k0 = koff;
        float2 av = A2[((size_t)mrow * D_ + k0) >> 1];
        v2f a; a.x = av.x - ((koff == 0) ? 1.0f : 0.0f); a.y = av.y;
        v2f bf;
        bf.x = Vb[(size_t)k0 * R_ + nn] * bmask;
        bf.y = Vb[(size_t)(k0 + 1) * R_ + nn] * bmask;
        c = wmma4(a, bf, c);
    }
    for (int kk = 4; kk < D_; kk += 4) {
        int k0 = kk + koff;
        float2 av = A2[((size_t)mrow * D_ + k0) >> 1];
        v2f a; a.x = av.x; a.y = av.y;
        v2f bf;
        bf.x = Vb[(size_t)k0 * R_ + nn] * bmask;
        bf.y = Vb[(size_t)(k0 + 1) * R_ + nn] * bmask;
        c = wmma4(a, bf, c);
    }
    if (n < R_) {
        int mb = (lane >> 4) << 3;
        for (int v = 0; v < 8; ++v)
            Y[((size_t)b * T_ + mt * 16 + mb + v) * R_ + n] = c[v];
    }
}

// Komega_apply on Y (real) -> Yout (real). One block per (b,j); loops 8 alphas.
__global__ void komega(const float* __restrict__ Yin, const float* __restrict__ w,
                       const float2* __restrict__ H, float* __restrict__ Yout,
                       FrftPack pk) {
    __shared__ float2 buf[L_];
    __shared__ float2 tmp[T_];
    __shared__ float yin[T_];
    __shared__ float acc[T_];
    int b = blockIdx.x / R_, j = blockIdx.x % R_;
    const float dt = 2.0f / 1023.0f;
    for (int t = threadIdx.x; t < T_; t += blockDim.x) {
        yin[t] = Yin[((size_t)b * T_ + t) * R_ + j];
        acc[t] = 0.f;
    }
    __syncthreads();
    for (int ai = 0; ai < 8; ++ai) {
        FrftP Pp = pk.p[ai];
        FrftP Pm = pk.p[8 + ai];
        const float* wrow = w + ((size_t)ai * B_ + b) * T_;
        for (int t = threadIdx.x; t < L_; t += blockDim.x) {
            float2 v = make_float2(0.f, 0.f);
            if (t < T_) {
                float2 pp = prepost_ph(Pp.cps, t);
                v = make_float2(yin[t] * pp.x, yin[t] * pp.y);
            }
            buf[t] = v;
        }
        fft2048(buf, -1.0f);
        for (int k = threadIdx.x; k < L_; k += blockDim.x)
            buf[k] = cmulf(buf[k], H[(size_t)ai * L_ + k]);
        fft2048(buf, +1.0f);
        float2 prefp = make_float2(Pp.prefRe, Pp.prefIm);
        for (int t = threadIdx.x; t < T_; t += blockDim.x) {
            float2 u = buf[1023 + t];
            u.x *= (1.0f / (float)L_); u.y *= (1.0f / (float)L_);
            float2 pp = prepost_ph(Pp.cps, t);
            float2 val = cmulf(cmulf(prefp, pp), u);
            float ww = wrow[t] * dt;
            tmp[t] = make_float2(val.x * ww, val.y * ww);
        }
        __syncthreads();
        for (int t = threadIdx.x; t < L_; t += blockDim.x) {
            float2 v = make_float2(0.f, 0.f);
            if (t < T_) {
                float2 pp = prepost_ph(Pm.cps, t);
                v = cmulf(tmp[t], pp);
            }
            buf[t] = v;
        }
        fft2048(buf, -1.0f);
        for (int k = threadIdx.x; k < L_; k += blockDim.x)
            buf[k] = cmulf(buf[k], H[(size_t)(8 + ai) * L_ + k]);
        fft2048(buf, +1.0f);
        float2 prefm = make_float2(Pm.prefRe, Pm.prefIm);
        for (int t = threadIdx.x; t < T_; t += blockDim.x) {
            float2 u = buf[1023 + t];
            u.x *= (1.0f / (float)L_); u.y *= (1.0f / (float)L_);
            float2 pp = prepost_ph(Pm.cps, t);
            float2 val = cmulf(cmulf(prefm, pp), u);
            acc[t] += val.x * dt;
        }
        __syncthreads();
    }
    for (int t = threadIdx.x; t < T_; t += blockDim.x)
        Yout[((size_t)b * T_ + t) * R_ + j] = acc[t] * 0.125f;
}

// Z[b,d,j] = (1/T) * sum_t xprime[b,t,d] * Y[b,t,j]   (WMMA, one wave per 16-d tile)
__global__ void gemm_zt(const float* __restrict__ x, const float* __restrict__ Y,
                        float* __restrict__ Z) {
    int b = blockIdx.x / (D_ / 16), mt = blockIdx.x % (D_ / 16);
    int lane = threadIdx.x;
    int mrow = lane & 15;
    int koff = (lane >> 4) << 1;
    int n = lane & 15;
    int nn = n & 7;
    float bmask = (n < R_) ? 1.0f : 0.0f;
    int dg = mt * 16 + mrow;
    float sub = (dg == 0) ? 1.0f : 0.0f;   // anchor column, hoisted out of the loop
    v8f c = {0.f, 0.f, 0.f, 0.f, 0.f, 0.f, 0.f, 0.f};
    for (int kk = 0; kk < T_; kk += 4) {
        int k0 = kk + koff;
        v2f a;
        a.x = x[((size_t)b * T_ + k0) * D_ + dg] - sub;
        a.y = x[((size_t)b * T_ + k0 + 1) * D_ + dg] - sub;
        const float2* Y2 = (const float2*)(Y + ((size_t)b * T_ + k0) * R_);
        v2f bf;
        bf.x = Y[((size_t)b * T_ + k0) * R_ + nn] * bmask;
        bf.y = Y[((size_t)b * T_ + k0 + 1) * R_ + nn] * bmask;
        (void)Y2;
        c = wmma4(a, bf, c);
    }
    if (n < R_) {
        int mb = (lane >> 4) << 3;
        for (int v = 0; v < 8; ++v)
            Z[((size_t)b * D_ + mt * 16 + mb + v) * R_ + n] = c[v] * (1.0f / (float)T_);
    }
}

// V = MGS-orthonormalize(Z + eps*V), per batch (matches QR + sign(diag R) convention)
__global__ void orthonorm(const float* __restrict__ Z, float* __restrict__ V) {
    __shared__ float Q[R_ * D_];
    __shared__ float red[256];
    int b = blockIdx.x;
    for (int i = threadIdx.x; i < D_ * R_; i += blockDim.x) {
        int d = i / R_, j = i % R_;
        size_t g = ((size_t)b * D_ + d) * R_ + j;
        Q[j * D_ + d] = Z[g] + EPSM * V[g];
    }
    __syncthreads();
    for (int j = 0; j < R_; ++j) {
        for (int i = 0; i < j; ++i) {
            float part = 0.f;
            for (int d = threadIdx.x; d < D_; d += blockDim.x)
                part += Q[i * D_ + d] * Q[j * D_ + d];
            red[threadIdx.x] = part;
            __syncthreads();
            for (int s = 128; s > 0; s >>= 1) {
                if (threadIdx.x < s) red[threadIdx.x] += red[threadIdx.x + s];
                __syncthreads();
            }
            float dot = red[0];
            __syncthreads();
            for (int d = threadIdx.x; d < D_; d += blockDim.x)
                Q[j * D_ + d] -= dot * Q[i * D_ + d];
            __syncthreads();
        }
        float part = 0.f;
        for (int d = threadIdx.x; d < D_; d += blockDim.x)
            part += Q[j * D_ + d] * Q[j * D_ + d];
        red[threadIdx.x] = part;
        __syncthreads();
        for (int s = 128; s > 0; s >>= 1) {
            if (threadIdx.x < s) red[threadIdx.x] += red[threadIdx.x + s];
            __syncthreads();
        }
        float inv = rsqrtf(fmaxf(red[0], 1e-30f));
        __syncthreads();
        for (int d = threadIdx.x; d < D_; d += blockDim.x) Q[j * D_ + d] *= inv;
        __syncthreads();
    }
    for (int i = threadIdx.x; i < D_ * R_; i += blockDim.x) {
        int d = i / R_, j = i % R_;
        V[((size_t)b * D_ + d) * R_ + j] = Q[j * D_ + d];
    }
}

__global__ void calc_signs(const float* __restrict__ V, float* __restrict__ signs) {
    int i = threadIdx.x;
    if (i < B_ * R_) {
        int b = i / R_, j = i % R_;
        float v = V[(size_t)b * D_ * R_ + j] + 1e-12f;
        signs[i] = (v > 0.f) ? 1.f : ((v < 0.f) ? -1.f : 0.f);
    }
}

__global__ void apply_signs(float* __restrict__ V, const float* __restrict__ signs) {
    int i = blockIdx.x * blockDim.x + threadIdx.x;
    if (i < B_ * D_ * R_) {
        int j = i % R_;
        int b = i / (D_ * R_);
        V[i] *= signs[b * R_ + j];
    }
}

__global__ void calc_scales(const float* __restrict__ tr, float* __restrict__ scales) {
    __shared__ float red[256];
    int b = blockIdx.x / R_, j = blockIdx.x % R_;
    float part = 0.f;
    for (int t = threadIdx.x; t < T_; t += blockDim.x) {
        float v = tr[((size_t)b * T_ + t) * R_ + j];
        part += v * v;
    }
    red[threadIdx.x] = part;
    __syncthreads();
    for (int s = 128; s > 0; s >>= 1) {
        if (threadIdx.x < s) red[threadIdx.x] += red[threadIdx.x + s];
        __syncthreads();
    }
    if (threadIdx.x == 0) scales[blockIdx.x] = sqrtf(red[0] + EPSM);
}

// soft-shrink with exact gelu, then re-scale (in place)
__global__ void shrink(float* __restrict__ tr, const float* __restrict__ scales) {
    int i = blockIdx.x * blockDim.x + threadIdx.x;
    if (i < B_ * T_ * R_) {
        int j = i % R_;
        int b = i / (T_ * R_);
        float sc = scales[b * R_ + j];
        float tn = tr[i] / sc;
        float a = fabsf(tn) - LAM_;
        float g = 0.5f * a * (1.0f + erff(a * 0.70710678118654752f));
        float sgn = (float)((tn > 0.f) - (tn < 0.f));
        tr[i] = sgn * g * sc;
    }
}

// x_hat[b,t,d] = sum_j tr[b,t,j] * V[b,d,j]  + anchor(d==0)   (WMMA, K=8)
__global__ void gemm_xtilde(const float* __restrict__ tr, const float* __restrict__ V,
                            float* __restrict__ xhat) {
    int blk = blockIdx.x;
    int b = blk / ((T_ / 16) * (D_ / 16));
    int rem = blk % ((T_ / 16) * (D_ / 16));
    int mt = rem / (D_ / 16), nt = rem % (D_ / 16);
    int lane = threadIdx.x;
    int mrow = lane & 15;
    int koff = (lane >> 4) << 1;
    int n = lane & 15;
    int dg = nt * 16 + n;
    const float2* A2 = (const float2*)(tr + ((size_t)b * T_ + mt * 16 + mrow) * R_);
    const float2* B2 = (const float2*)(V + ((size_t)b * D_ + dg) * R_);
    v8f c = {0.f, 0.f, 0.f, 0.f, 0.f, 0.f, 0.f, 0.f};
    for (int kk = 0; kk < R_; kk += 4) {
        int k0 = kk + koff;
        float2 av = A2[k0 >> 1];
        float2 bv = B2[k0 >> 1];
        v2f a; a.x = av.x; a.y = av.y;
        v2f bf; bf.x = bv.x; bf.y = bv.y;
        c = wmma4(a, bf, c);
    }
    int mb = (lane >> 4) << 3;
    float anc = (dg == 0) ? 1.0f : 0.0f;
    for (int v = 0; v < 8; ++v)
        xhat[((size_t)b * T_ + mt * 16 + mb + v) * D_ + dg] = c[v] + anc;
}

// Fused: y = x + xhat @ W^T, then LayerNorm(D). 16 rows/block.
// A-panel (16x512 f32 = contiguous 32KB of xhat) staged into LDS via the
// Tensor Data Mover (1-D D#, data_size=4B, tile_dim0=8192 elements).
__global__ void __launch_bounds__(256)
gemm_out_ln(const float* __restrict__ xhat, const float* __restrict__ W,
            const float* __restrict__ x, const float* __restrict__ gamma,
            const float* __restrict__ beta, float* __restrict__ out) {
    __shared__ float sA[16 * D_];   // 32 KB (TDM destination)
    __shared__ float sC[16 * D_];   // 32 KB
    int row0 = blockIdx.x * 16;     // global row in flattened [B*T]

    if (threadIdx.x < 32) {         // wave 0 drives the TDM
        unsigned lds_off = (unsigned)(unsigned long long)(uintptr_t)(void*)sA;
        unsigned long long ga =
            (unsigned long long)(uintptr_t)(xhat + (size_t)row0 * D_);
        v4i g0;
        g0[0] = 1;                                            // count=1, user mode
        g0[1] = (int)lds_off;                                 // lds_addr
        g0[2] = (int)(unsigned)(ga & 0xffffffffu);            // global_addr[31:0]
        g0[3] = (int)((((unsigned)(ga >> 32)) & 0x01ffffffu)  // global_addr[56:32]
                      | 0x80000000u);                         // type=2 (image)
        v8i g1;
        g1[0] = 0x00020000;            // workgroup_mask=0, data_size=2 (4 bytes)
        g1[1] = (int)(8192u << 16);    // tensor_dim0[15:0]=8192 (in bits 31:16)
        g1[2] = (int)((8192u >> 16) & 0xffffu) | (1 << 16);  // dim0 hi, tensor_dim1=1
        g1[3] = (int)(8192u << 16);    // tensor_dim1 hi=0, tile_dim0=8192
        g1[4] = 0;                     // tile_dim1=0 (unused), tile_dim2=0
        g1[5] = 8192;                  // tensor_dim0_stride[31:0]
        g1[6] = 0;                     // stride0 hi, tensor_dim1_stride lo
        g1[7] = 0;
        asm volatile("tensor_load_to_lds %0, %1"
                     :: "s"(g0), "s"(g1)
                     : "memory");
        __builtin_amdgcn_s_wait_tensorcnt(0);
    }
    __syncthreads();

    int wave = threadIdx.x >> 5;
    int lane = threadIdx.x & 31;
    int mrow = lane & 15;
    int koff = (lane >> 4) << 1;
    int n = lane & 15;
    const float2* sA2 = (const float2*)sA;
    for (int q = 0; q < 4; ++q) {
        int nt = wave * 4 + q;
        const float2* Wn2 = (const float2*)(W + (size_t)(nt * 16 + n) * D_);
        __builtin_prefetch(Wn2, 0, 3);
        v8f c = {0.f, 0.f, 0.f, 0.f, 0.f, 0.f, 0.f, 0.f};
        for (int kk = 0; kk < D_; kk += 4) {
            int k0 = kk + koff;
            float2 av = sA2[(mrow * D_ + k0) >> 1];
            float2 bv = Wn2[k0 >> 1];
            v2f a; a.x = av.x; a.y = av.y;
            v2f bf; bf.x = bv.x; bf.y = bv.y;
            c = wmma4(a, bf, c);
        }
        int mb = (lane >> 4) << 3;
        for (int v = 0; v < 8; ++v) {
            int m = mb + v;
            int col = nt * 16 + n;
            sC[m * D_ + col] = c[v] + x[(size_t)(row0 + m) * D_ + col];
        }
    }
    __syncthreads();
    for (int rr = 0; rr < 2; ++rr) {
        int m = wave * 2 + rr;
        float sum = 0.f, sumsq = 0.f;
        for (int d = lane; d < D_; d += 32) {
            float v = sC[m * D_ + d];
            sum += v; sumsq += v * v;
        }
        for (int off = 16; off > 0; off >>= 1) {
            sum += __shfl_xor(sum, off, 32);
            sumsq += __shfl_xor(sumsq, off, 32);
        }
        float mu = sum * (1.0f / (float)D_);
        float var = sumsq * (1.0f / (float)D_) - mu * mu;
        float rstd = rsqrtf(var + 1e-5f);
        for (int d = lane; d < D_; d += 32) {
            float v = (sC[m * D_ + d] - mu) * rstd * gamma[d] + beta[d];
            out[(size_t)(row0 + m) * D_ + d] = v;
        }
    }
}

// ---------------- host ----------------
static void make_params(FrftPack& pk) {
    const double pi = 3.14159265358979323846;
    const double eps = 1e-7;
    for (int k = 0; k < 16; ++k) {
        double alpha0 = 0.15 + (2.99 - 0.15) * (double)(k & 7) / 7.0;
        double alpha = (k < 8) ? alpha0 : -alpha0;
        double a = fmod(alpha + pi, 2.0 * pi);
        if (a < 0.0) a += 2.0 * pi;
        a -= pi;
        double sin_a = sin(a);
        double sgn = (sin_a > 0.0) ? 1.0 : ((sin_a < 0.0) ? -1.0 : 0.0);
        double s = sgn / fmax(eps, fabs(sin_a));
        double c = cos(a) / fmax(eps, sin_a);   // faithful signed-sin clamp
        std::complex<double> pref = std::sqrt(std::complex<double>(1.0, -c));
        pk.p[k].s = (float)s;
        pk.p[k].cps = (float)(c + s);
        pk.p[k].prefRe = (float)pref.real();
        pk.p[k].prefIm = (float)pref.imag();
    }
}

extern "C" void kernel_launch(void* const* d_in, const int* in_sizes, int n_in,
                              void* d_out, int out_size, void* d_ws, size_t ws_size,
                              hipStream_t stream) {
    const float* x = (const float*)d_in[0];
    const float* W = (const float*)d_in[1];
    const float* gamma = (const float*)d_in[2];
    const float* beta = (const float*)d_in[3];
    float* out = (float*)d_out;

    char* ws = (char*)d_ws;
    size_t o = 0;
    float2* H = (float2*)(ws + o);  o += (size_t)16 * L_ * sizeof(float2);      // 256 KB
    float* Ew = (float*)(ws + o);   o += (size_t)8 * B_ * T_ * sizeof(float);   // 128 KB
    float* V = (float*)(ws + o);    o += (size_t)B_ * D_ * R_ * sizeof(float);
    float* Z = (float*)(ws + o);    o += (size_t)B_ * D_ * R_ * sizeof(float);
    float* Ybuf = (float*)(ws + o); o += (size_t)B_ * T_ * R_ * sizeof(float);
    float* Yout = (float*)(ws + o); o += (size_t)B_ * T_ * R_ * sizeof(float);
    float* traces = (float*)(ws + o); o += (size_t)B_ * T_ * R_ * sizeof(float);
    float* scales = (float*)(ws + o); o += 256;
    float* signs = (float*)(ws + o);  o += 256;
    float* xhat = (float*)(ws + o);   o += (size_t)B_ * T_ * D_ * sizeof(float); // 8 MB

    FrftPack pk;
    make_params(pk);

    const int EWN = 8 * B_ * T_;
    zero_f32<<<(EWN + 255) / 256, 256, 0, stream>>>(Ew, EWN);
    build_H<<<16, 256, 0, stream>>>(H, pk);
    frft_energy<<<8 * B_ * D_, 256, 0, stream>>>(x, H, Ew, pk);
    finalize_w<<<8 * B_, 256, 0, stream>>>(Ew);

    const int BDR = B_ * D_ * R_;
    init_V<<<(BDR + 255) / 256, 256, 0, stream>>>(V);

    for (int it = 0; it < 2; ++it) {
        gemm_xprime_V<<<B_ * (T_ / 16), 32, 0, stream>>>(x, V, Ybuf);
        komega<<<B_ * R_, 256, 0, stream>>>(Ybuf, Ew, H, Yout, pk);
        gemm_zt<<<B_ * (D_ / 16), 32, 0, stream>>>(x, Yout, Z);
        orthonorm<<<B_, 256, 0, stream>>>(Z, V);
    }

    calc_signs<<<1, 32, 0, stream>>>(V, signs);
    apply_signs<<<(BDR + 255) / 256, 256, 0, stream>>>(V, signs);

    gemm_xprime_V<<<B_ * (T_ / 16), 32, 0, stream>>>(x, V, traces);
    calc_scales<<<B_ * R_, 256, 0, stream>>>(traces, scales);
    const int BTR = B_ * T_ * R_;
    shrink<<<(BTR + 255) / 256, 256, 0, stream>>>(traces, scales);

    gemm_xtilde<<<B_ * (T_ / 16) * (D_ / 16), 32, 0, stream>>>(traces, V, xhat);
    gemm_out_ln<<<(B_ * T_) / 16, 256, 0, stream>>>(xhat, W, x, gamma, beta, out);

    (void)in_sizes; (void)n_in; (void)out_size; (void)ws_size;
}